// MetricsLoss_65781719106339
// MI455X (gfx1250) — compile-verified
//
#include <hip/hip_runtime.h>
#include <stdint.h>

typedef __attribute__((ext_vector_type(2))) float v2f;
typedef __attribute__((ext_vector_type(8))) float v8f;

#define TILE_ROWS 16
#define ROW_PAD   132      // 128 floats + 4 pad: float4-aligned rows, bank-conflict-free fragment reads
#define WAVES1    4
#define NBLK1     2048
#define S_GRP     128
#define MARGIN_F  2.0f
#define K_MARGIN  0.02f
#define EPS_F     1e-6f

// ---------------------------------------------------------------------------
// Kernel 1: d[i] = 1 - dot(z_r[i], z_v[i]).  Memory-bound (512 MB streamed).
// Async global->LDS streaming + f32 WMMA (diagonal of A * B^T gives row dots).
// Also produces per-block partial sums for the label-conditional means.
// ---------------------------------------------------------------------------
__global__ __launch_bounds__(128) void mloss_dot_kernel(
    const float* __restrict__ zr, const float* __restrict__ zv,
    const int* __restrict__ labels, float* __restrict__ d_vec,
    float* __restrict__ part, int nTiles, int nBlocks)
{
  __shared__ float lds[WAVES1][2][TILE_ROWS * ROW_PAD];
  __shared__ float wB[WAVES1], wP[WAVES1], wC[WAVES1];

  const int lane = threadIdx.x & 31;
  const int warp = threadIdx.x >> 5;

  float sum_b = 0.f, sum_p = 0.f, cnt_b = 0.f;

  const unsigned int ldsR = (unsigned int)(uintptr_t)&lds[warp][0][0];
  const unsigned int ldsV = (unsigned int)(uintptr_t)&lds[warp][1][0];

  // WMMA A/B fragment read positions for this lane:
  //   lanes 0-15  hold (M=lane,   K=kk+0 / kk+1) in VGPR0/1
  //   lanes 16-31 hold (M=lane-16,K=kk+2 / kk+3) in VGPR0/1
  const int m    = lane & 15;
  const int koff = (lane < 16) ? 0 : 2;
  const float* rdR = &lds[warp][0][m * ROW_PAD + koff];
  const float* rdV = &lds[warp][1][m * ROW_PAD + koff];

  for (int tile = blockIdx.x * WAVES1 + warp; tile < nTiles;
       tile += nBlocks * WAVES1) {
    const long long row0 = (long long)tile * TILE_ROWS;

    // make sure previous iteration's DS traffic retired before LDS is overwritten
    asm volatile("s_wait_dscnt 0x0" ::: "memory");

#pragma unroll
    for (int r = 0; r < TILE_ROWS; ++r) {
      unsigned long long gr =
          (unsigned long long)(uintptr_t)(zr + (row0 + r) * 128 + lane * 4);
      unsigned long long gv =
          (unsigned long long)(uintptr_t)(zv + (row0 + r) * 128 + lane * 4);
      unsigned int lr = ldsR + (unsigned int)(r * ROW_PAD + lane * 4) * 4u;
      unsigned int lv = ldsV + (unsigned int)(r * ROW_PAD + lane * 4) * 4u;
      // one instruction = one fully coalesced 512B row -> LDS, tracked by ASYNCcnt
      asm volatile("global_load_async_to_lds_b128 %0, %1, off"
                   :: "v"(lr), "v"(gr) : "memory");
      asm volatile("global_load_async_to_lds_b128 %0, %1, off"
                   :: "v"(lv), "v"(gv) : "memory");
    }
    asm volatile("s_wait_asynccnt 0x0" ::: "memory");

    float dotv = 0.f;
#if __has_builtin(__builtin_amdgcn_wmma_f32_16x16x4_f32)
    // Accumulate C[m][n] = sum_k z_r[m,k] * z_v[n,k]; diagonal = row dots.
    v8f c = {0.f, 0.f, 0.f, 0.f, 0.f, 0.f, 0.f, 0.f};
#pragma unroll
    for (int kk = 0; kk < 128; kk += 4) {
      v2f a = { rdR[kk], rdR[kk + 1] };
      v2f b = { rdV[kk], rdV[kk + 1] };
      c = __builtin_amdgcn_wmma_f32_16x16x4_f32(false, a, false, b,
                                                (short)0, c, false, false);
    }
    // Stage C to LDS (reuse tile-R region) and pull the diagonal.
    // C/D layout: (M<8) -> VGPR M, lane N ; (M>=8) -> VGPR M-8, lane 16+N.
    float* cst = &lds[warp][0][0];
#pragma unroll
    for (int r = 0; r < 8; ++r) cst[r * 32 + lane] = c[r];
    asm volatile("" ::: "memory");   // per-wave DS ops complete in order
    if (lane < 16) {
      dotv = (lane < 8) ? cst[lane * 32 + lane]
                        : cst[(lane - 8) * 32 + 16 + lane];
    }
#else
    if (lane < 16) {
      const float* ar = &lds[warp][0][lane * ROW_PAD];
      const float* av = &lds[warp][1][lane * ROW_PAD];
      for (int k = 0; k < 128; ++k) dotv += ar[k] * av[k];
    }
#endif

    if (lane < 16) {
      const float dv = 1.0f - dotv;
      const long long row = row0 + lane;
      d_vec[row] = dv;
      if (labels[row] == 0) { sum_b += dv; cnt_b += 1.f; }
      else                  { sum_p += dv; }
    }
  }

  // wave reduction (fixed xor pattern -> deterministic)
#pragma unroll
  for (int o = 16; o >= 1; o >>= 1) {
    sum_b += __shfl_xor(sum_b, o, 32);
    sum_p += __shfl_xor(sum_p, o, 32);
    cnt_b += __shfl_xor(cnt_b, o, 32);
  }
  if (lane == 0) { wB[warp] = sum_b; wP[warp] = sum_p; wC[warp] = cnt_b; }
  __syncthreads();
  if (threadIdx.x == 0) {
    float b = 0.f, p = 0.f, cc = 0.f;
    for (int w = 0; w < WAVES1; ++w) { b += wB[w]; p += wP[w]; cc += wC[w]; }
    part[blockIdx.x * 3 + 0] = b;
    part[blockIdx.x * 3 + 1] = p;
    part[blockIdx.x * 3 + 2] = cc;
  }
}

// ---------------------------------------------------------------------------
// Kernel 2: per-group losses. One 128-thread block per group of S=128.
// ---------------------------------------------------------------------------
__global__ __launch_bounds__(128) void mloss_group_kernel(
    const float* __restrict__ d_vec, const int* __restrict__ var_lens,
    float* __restrict__ gloss)
{
  __shared__ float red[S_GRP];
  __shared__ int   svlen[S_GRP];
  __shared__ float dsrt[S_GRP];
  __shared__ int   vsrt[S_GRP];

  const int t = threadIdx.x;
  const long long base = (long long)blockIdx.x * S_GRP;
  const float dd = d_vec[base + t];
  const int   vi = var_lens[base + t];
  const float v  = (float)vi;
  svlen[t] = vi;
  __syncthreads();

  auto reduce = [&](float val) -> float {
    red[t] = val; __syncthreads();
    for (int o = 64; o >= 1; o >>= 1) {
      if (t < o) red[t] += red[t + o];
      __syncthreads();
    }
    float r = red[0]; __syncthreads();
    return r;
  };

  const float mv = reduce(v)  * (1.f / S_GRP);
  const float md = reduce(dd) * (1.f / S_GRP);
  const float vs = sqrtf(reduce((v  - mv) * (v  - mv)) / (float)(S_GRP - 1));
  const float ds = sqrtf(reduce((dd - md) * (dd - md)) / (float)(S_GRP - 1));
  const float vz = (v  - mv) / (vs + EPS_F);
  const float dz = (dd - md) / (ds + EPS_F);
  float corr = reduce((vz - dz) * (vz - dz)) * (1.f / S_GRP);
  if (!((vs > 0.f) && (ds > 0.f))) corr = 0.f;

  // stable counting rank == jnp.argsort (stable) position
  int rank = 0;
  for (int j = 0; j < S_GRP; ++j) {
    const int vj = svlen[j];
    rank += (int)((vj < vi) || ((vj == vi) && (j < t)));
  }
  vsrt[rank] = vi;
  dsrt[rank] = dd;
  __syncthreads();

  float nv = 0.f;
  if (t < S_GRP - 1) nv = fmaxf(0.f, dsrt[t] - dsrt[t + 1] + K_MARGIN);
  const float neigh = reduce(nv) * (1.f / (float)(S_GRP - 1));

  // pairwise rank loss: dv[i,j] = v[j]-v[i] > 0 ; viol = relu(K - (d[j]-d[i]))
  float rs = 0.f, rc = 0.f;
  const int   vi_s = vsrt[t];
  const float dd_s = dsrt[t];
  for (int j = 0; j < S_GRP; ++j) {
    if (vsrt[j] > vi_s) {
      rc += 1.f;
      rs += fmaxf(0.f, K_MARGIN - (dsrt[j] - dd_s));
    }
  }
  const float rsum = reduce(rs);
  const float rcnt = reduce(rc);
  const float rank_loss = (rcnt > 0.f) ? (rsum / rcnt) : 0.f;

  if (t == 0) gloss[blockIdx.x] = corr + neigh + rank_loss;
}

// ---------------------------------------------------------------------------
// Kernel 3: deterministic final reductions -> l_cdd, l_pcc.
// ---------------------------------------------------------------------------
__global__ __launch_bounds__(256) void mloss_final_kernel(
    const float* __restrict__ part, int nPart,
    const float* __restrict__ gloss, int nGroups,
    float* __restrict__ out, float nTotal)
{
  __shared__ float s0[256], s1[256], s2[256], s3[256];
  const int t = threadIdx.x;
  float sb = 0.f, sp = 0.f, cb = 0.f, gl = 0.f;
  for (int i = t; i < nPart; i += 256) {
    sb += part[3 * i + 0];
    sp += part[3 * i + 1];
    cb += part[3 * i + 2];
  }
  for (int i = t; i < nGroups; i += 256) gl += gloss[i];
  s0[t] = sb; s1[t] = sp; s2[t] = cb; s3[t] = gl;
  __syncthreads();
  for (int o = 128; o >= 1; o >>= 1) {
    if (t < o) {
      s0[t] += s0[t + o]; s1[t] += s1[t + o];
      s2[t] += s2[t + o]; s3[t] += s3[t + o];
    }
    __syncthreads();
  }
  if (t == 0) {
    const float nb = s2[0];
    const float np = nTotal - s2[0];
    const float mean_b = s0[0] / fmaxf(nb, 1.f);
    const float mean_p = s1[0] / fmaxf(np, 1.f);
    const float l_cdd =
        (nb > 0.f && np > 0.f) ? fmaxf(0.f, MARGIN_F + mean_b - mean_p) : 0.f;
    out[0] = l_cdd;
    out[1] = s3[0] / (float)nGroups;   // uniform group weights = 1/G
  }
}

// ---------------------------------------------------------------------------
extern "C" void kernel_launch(void* const* d_in, const int* in_sizes, int n_in,
                              void* d_out, int out_size, void* d_ws, size_t ws_size,
                              hipStream_t stream) {
  const float* zr       = (const float*)d_in[0];
  const float* zv       = (const float*)d_in[1];
  const int*   labels   = (const int*)d_in[2];
  const int*   var_lens = (const int*)d_in[4];   // d_in[3] = groups (implied by layout)

  const int N      = in_sizes[2];        // number of rows
  const int G      = N / S_GRP;
  const int nTiles = N / TILE_ROWS;

  float* out   = (float*)d_out;          // [0]=l_cdd, [1]=l_pcc, [2..2+N) = d
  float* d_vec = out + 2;
  float* part  = (float*)d_ws;           // 3*NBLK1 floats
  float* gloss = part + 3 * NBLK1;       // G floats

  mloss_dot_kernel<<<NBLK1, 128, 0, stream>>>(zr, zv, labels, d_vec, part,
                                              nTiles, NBLK1);
  mloss_group_kernel<<<G, 128, 0, stream>>>(d_vec, var_lens, gloss);
  mloss_final_kernel<<<1, 256, 0, stream>>>(part, NBLK1, gloss, G, out,
                                            (float)N);
}